// GATConv_56908316672598
// MI455X (gfx1250) — compile-verified
//
#include <hip/hip_runtime.h>
#include <math.h>

#define N_NODES 50000
#define N_EDGES 1600000
#define DIM_IN  256
#define HEADS   4
#define DIM_OUT 32
#define DIM_HID 128   /* HEADS*DIM_OUT */
#define NEG_SLOPE 0.2f

typedef float v2f __attribute__((ext_vector_type(2)));
typedef float v8f __attribute__((ext_vector_type(8)));

// Order-preserving float->uint encoding so atomicMax(u32) == float max.
__device__ __forceinline__ unsigned enc_f32(float f) {
    unsigned u = __float_as_uint(f);
    return (u & 0x80000000u) ? ~u : (u | 0x80000000u);
}
__device__ __forceinline__ float dec_f32(unsigned u) {
    unsigned b = (u & 0x80000000u) ? (u & 0x7FFFFFFFu) : ~u;
    return __uint_as_float(b);
}
__device__ __forceinline__ float lrelu(float s) {
    return s > 0.0f ? s : NEG_SLOPE * s;
}

// ---------------------------------------------------------------------------
// Kernel 1: xp = x @ W via V_WMMA_F32_16X16X4_F32.
// Block = 256 threads = 8 waves; each wave computes one 16x16 tile; the block
// covers 16 rows x 128 cols. Grid = N/16 = 3125 blocks (exact).
// A-frag: lane m=lane&15 holds K = 2*(lane>>4)+{0,1}  (contiguous float2).
// B-frag: lane n=lane&15 holds rows K = 2*(lane>>4)+{0,1} of W.
// D: vgpr r -> element [rowbase + r + 8*(lane>>4)][colbase + (lane&15)].
// ---------------------------------------------------------------------------
__global__ __launch_bounds__(256) void gat_gemm_wmma(
        const float* __restrict__ x, const float* __restrict__ W,
        float* __restrict__ xp) {
    const int lane = threadIdx.x & 31;
    const int wave = threadIdx.x >> 5;
    const int row_base = blockIdx.x * 16;
    const int col_base = wave * 16;
    const int m = lane & 15;
    const int khalf = (lane >> 4) * 2;   // 0 or 2

    const float* xrow = x + (size_t)(row_base + m) * DIM_IN;
    v8f acc = {};
    for (int k = 0; k < DIM_IN; k += 4) {
        v2f a = *(const v2f*)(xrow + k + khalf);               // 8B-aligned
        v2f b;
        b.x = W[(size_t)(k + khalf + 0) * DIM_HID + col_base + m];
        b.y = W[(size_t)(k + khalf + 1) * DIM_HID + col_base + m];
        acc = __builtin_amdgcn_wmma_f32_16x16x4_f32(
            /*neg_a=*/false, a, /*neg_b=*/false, b,
            /*c_mod=*/(short)0, acc, /*reuse_a=*/false, /*reuse_b=*/false);
    }
    const int rrow = row_base + 8 * (lane >> 4);
#pragma unroll
    for (int r = 0; r < 8; ++r)
        xp[(size_t)(rrow + r) * DIM_HID + col_base + m] = acc[r];
}

// ---------------------------------------------------------------------------
// Kernel 2: per-(node,head) attention logits a_src/a_dst (dot over 32 chans).
// ---------------------------------------------------------------------------
__global__ void gat_logits(const float* __restrict__ xp,
                           const float* __restrict__ att_src,
                           const float* __restrict__ att_dst,
                           float* __restrict__ a_src,
                           float* __restrict__ a_dst) {
    int idx = blockIdx.x * blockDim.x + threadIdx.x;   // n*HEADS + h
    if (idx >= N_NODES * HEADS) return;
    const int h = idx & (HEADS - 1);
    const float* v  = xp + (size_t)(idx >> 2) * DIM_HID + h * DIM_OUT;
    const float* as = att_src + h * DIM_OUT;
    const float* ad = att_dst + h * DIM_OUT;
    float ssum = 0.0f, dsum = 0.0f;
#pragma unroll
    for (int c = 0; c < DIM_OUT; ++c) {
        float xv = v[c];
        ssum = fmaf(xv, as[c], ssum);
        dsum = fmaf(xv, ad[c], dsum);
    }
    a_src[idx] = ssum;
    a_dst[idx] = dsum;
}

// ---------------------------------------------------------------------------
// Kernel 3: init out = bias (so atomics accumulate into the final answer),
// emax = 0 (encoded -inf floor; self-loops guarantee >=1 real score),
// denom = 0.
// ---------------------------------------------------------------------------
__global__ void gat_init(float* __restrict__ out, const float* __restrict__ bias,
                         unsigned* __restrict__ emax, float* __restrict__ denom) {
    int idx = blockIdx.x * blockDim.x + threadIdx.x;
    if (idx < N_NODES * DIM_HID) out[idx] = bias[idx & (DIM_HID - 1)];
    if (idx < N_NODES * HEADS) { emax[idx] = 0u; denom[idx] = 0.0f; }
}

__device__ __forceinline__ void edge_endpoints(const long long* __restrict__ ei,
                                               int e, int& src, int& dst) {
    if (e < N_EDGES) { src = (int)ei[e]; dst = (int)ei[(size_t)N_EDGES + e]; }
    else             { src = dst = e - N_EDGES; }          // self loops
}

// ---------------------------------------------------------------------------
// Kernel 4: segment max of leaky-relu scores per (dst, head).
// ---------------------------------------------------------------------------
__global__ void gat_segmax(const long long* __restrict__ ei,
                           const float* __restrict__ a_src,
                           const float* __restrict__ a_dst,
                           unsigned* __restrict__ emax) {
    int e = blockIdx.x * blockDim.x + threadIdx.x;
    if (e >= N_EDGES + N_NODES) return;
    int src, dst; edge_endpoints(ei, e, src, dst);
#pragma unroll
    for (int h = 0; h < HEADS; ++h) {
        float s = lrelu(a_src[src * HEADS + h] + a_dst[dst * HEADS + h]);
        atomicMax(&emax[dst * HEADS + h], enc_f32(s));
    }
}

// ---------------------------------------------------------------------------
// Kernel 5: segment sum of exp(e - emax) per (dst, head).
// ---------------------------------------------------------------------------
__global__ void gat_denom(const long long* __restrict__ ei,
                          const float* __restrict__ a_src,
                          const float* __restrict__ a_dst,
                          const unsigned* __restrict__ emax,
                          float* __restrict__ denom) {
    int e = blockIdx.x * blockDim.x + threadIdx.x;
    if (e >= N_EDGES + N_NODES) return;
    int src, dst; edge_endpoints(ei, e, src, dst);
#pragma unroll
    for (int h = 0; h < HEADS; ++h) {
        float s = lrelu(a_src[src * HEADS + h] + a_dst[dst * HEADS + h]);
        float ex = __expf(s - dec_f32(emax[dst * HEADS + h]));
        atomicAdd(&denom[dst * HEADS + h], ex);
    }
}

// ---------------------------------------------------------------------------
// Kernel 6: scatter-aggregate. One wave per edge: lane handles 4 channels
// (float4 gather from xp[src], 4 f32 atomics into out[dst]); L2-resident.
// ---------------------------------------------------------------------------
__global__ __launch_bounds__(256) void gat_aggregate(
        const long long* __restrict__ ei,
        const float* __restrict__ xp,
        const float* __restrict__ a_src,
        const float* __restrict__ a_dst,
        const unsigned* __restrict__ emax,
        const float* __restrict__ denom,
        float* __restrict__ out) {
    const int lane = threadIdx.x & 31;
    const int e = blockIdx.x * 8 + (threadIdx.x >> 5);
    if (e >= N_EDGES + N_NODES) return;
    int src, dst; edge_endpoints(ei, e, src, dst);

    const int h = lane >> 3;                 // 8 lanes per head (4*8=32 chans)
    float s = lrelu(a_src[src * HEADS + h] + a_dst[dst * HEADS + h]);
    float alpha = __expf(s - dec_f32(emax[dst * HEADS + h]))
                  / denom[dst * HEADS + h];

    const float4 v = *(const float4*)(xp + (size_t)src * DIM_HID + lane * 4);
    float* o = out + (size_t)dst * DIM_HID + lane * 4;
    atomicAdd(o + 0, v.x * alpha);
    atomicAdd(o + 1, v.y * alpha);
    atomicAdd(o + 2, v.z * alpha);
    atomicAdd(o + 3, v.w * alpha);
}

// ---------------------------------------------------------------------------
extern "C" void kernel_launch(void* const* d_in, const int* in_sizes, int n_in,
                              void* d_out, int out_size, void* d_ws, size_t ws_size,
                              hipStream_t stream) {
    const float*     x       = (const float*)d_in[0];
    const long long* ei      = (const long long*)d_in[1];
    const float*     W       = (const float*)d_in[2];
    const float*     att_src = (const float*)d_in[3];
    const float*     att_dst = (const float*)d_in[4];
    const float*     bias    = (const float*)d_in[5];
    float*           out     = (float*)d_out;

    char* ws = (char*)d_ws;
    float*    xp    = (float*)ws;    ws += (size_t)N_NODES * DIM_HID * sizeof(float);
    float*    a_src = (float*)ws;    ws += (size_t)N_NODES * HEADS * sizeof(float);
    float*    a_dst = (float*)ws;    ws += (size_t)N_NODES * HEADS * sizeof(float);
    unsigned* emax  = (unsigned*)ws; ws += (size_t)N_NODES * HEADS * sizeof(unsigned);
    float*    denom = (float*)ws;

    const int ET = N_EDGES + N_NODES;

    gat_gemm_wmma<<<N_NODES / 16, 256, 0, stream>>>(x, W, xp);
    gat_logits<<<(N_NODES * HEADS + 255) / 256, 256, 0, stream>>>(
        xp, att_src, att_dst, a_src, a_dst);
    gat_init<<<(N_NODES * DIM_HID + 255) / 256, 256, 0, stream>>>(
        out, bias, emax, denom);
    gat_segmax<<<(ET + 255) / 256, 256, 0, stream>>>(ei, a_src, a_dst, emax);
    gat_denom<<<(ET + 255) / 256, 256, 0, stream>>>(ei, a_src, a_dst, emax, denom);
    gat_aggregate<<<(ET + 7) / 8, 256, 0, stream>>>(
        ei, xp, a_src, a_dst, emax, denom, out);
}